// ChannelAttentionModule_9466107920581
// MI455X (gfx1250) — compile-verified
//
#include <hip/hip_runtime.h>
#include <hip/hip_bf16.h>
#include <math.h>

// ---------------------------------------------------------------------------
// ChannelAttentionModule for MI455X (gfx1250), wave32.
//  B=64, H=64, W=64, C=256, E=3 (hidden 768), windows=8.
//  Phase 1: fused streaming pass over x (256MB, HBM-bound):
//           per-(h,w,c) prefix sums over batch -> cyclic window means ->
//           running max (registers) ; spatial sums in LDS (avg path commutes).
//  Phase 2: tiny f32 GEMMs via v_wmma_f32_16x16x4_f32 (full f32 accuracy).
//  Phase 3: sigmoid + rowwise softmax + final cyclic window mean.
// ---------------------------------------------------------------------------

#define Bsz   64
#define CCH   256
#define HID   768
#define SPAT  4096            // H*W
#define WIN   8

typedef __attribute__((ext_vector_type(2))) float v2f;
typedef __attribute__((ext_vector_type(8))) float v8f;

// -------- order-preserving float <-> uint encoding for atomic max ----------
__device__ __forceinline__ unsigned enc_f32(float f) {
    unsigned u = __float_as_uint(f);
    return (u & 0x80000000u) ? ~u : (u | 0x80000000u);
}
__device__ __forceinline__ float dec_f32(unsigned k) {
    return (k & 0x80000000u) ? __uint_as_float(k ^ 0x80000000u)
                             : __uint_as_float(~k);
}

// encoded(-inf) = ~0xFF800000 = 0x007FFFFF
#define ENC_NEG_INF 0x007FFFFFu

// ---------------------------------------------------------------------------
__global__ void k_init(unsigned* __restrict__ mx_enc, float* __restrict__ av_sum) {
    int t = blockIdx.x * blockDim.x + threadIdx.x;
    if (t < Bsz * CCH) {
        mx_enc[t] = ENC_NEG_INF;
        av_sum[t] = 0.0f;
    }
}

// ---------------------------------------------------------------------------
// One block = 16 spatial locations x all 256 channels x all 64 batch rows.
// thread = channel. Grid = 256 blocks (4096/16 spatial chunks).
__global__ void k_window_pool(const float* __restrict__ x,
                              unsigned* __restrict__ mx_enc,
                              float* __restrict__ av_sum) {
    extern __shared__ float lds_av[];        // [64][256] spatial-sum accum
    const int c = threadIdx.x;

    #pragma unroll
    for (int b = 0; b < Bsz; ++b) lds_av[b * CCH + c] = 0.0f;

    float lmax[Bsz];
    #pragma unroll
    for (int i = 0; i < Bsz; ++i) lmax[i] = -3.402823466e38f;

    const int s0 = blockIdx.x * 16;
    for (int sl = 0; sl < 16; ++sl) {
        const int s = s0 + sl;
        const float* p = x + (size_t)s * CCH + c;
        float v[Bsz];                         // running prefix sums S[b+1]
        float pref = 0.0f;
        #pragma unroll
        for (int b = 0; b < Bsz; ++b) {
            float val = p[(size_t)b * (SPAT * CCH)];
            lds_av[b * CCH + c] += val;       // thread owns column c: race-free
            pref += val;
            v[b] = pref;
        }
        // cyclic window means from prefix sums; update running max
        #pragma unroll
        for (int i = 0; i < WIN; ++i) {       // wrap branch: (i+1)+(8-i)=9 rows
            float m = (v[i] + v[Bsz - 1] - v[55 + i]) * (1.0f / 9.0f);
            lmax[i] = fmaxf(lmax[i], m);
        }
        lmax[WIN] = fmaxf(lmax[WIN], v[WIN - 1] * 0.125f);   // i==8: S[8]-S[0]
        #pragma unroll
        for (int i = WIN + 1; i < Bsz; ++i) {
            float m = (v[i - 1] - v[i - 9]) * 0.125f;
            lmax[i] = fmaxf(lmax[i], m);
        }
    }

    // cross-block reductions
    #pragma unroll
    for (int i = 0; i < Bsz; ++i)
        atomicMax(&mx_enc[i * CCH + c], enc_f32(lmax[i]));   // order-invariant
    #pragma unroll
    for (int b = 0; b < Bsz; ++b)
        atomicAdd(&av_sum[b * CCH + c], lds_av[b * CCH + c]);
}

// ---------------------------------------------------------------------------
// grid 64 (i) x 256 (c): decode max path, window-mean the avg path.
__global__ void k_finalize(const unsigned* __restrict__ mx_enc,
                           const float* __restrict__ av_sum,
                           float* __restrict__ MX, float* __restrict__ AV) {
    const int c = threadIdx.x;
    const int i = blockIdx.x;
    MX[i * CCH + c] = dec_f32(mx_enc[i * CCH + c]);

    const float scale = 1.0f / (float)SPAT;
    float sum = 0.0f;
    if (i < WIN) {
        for (int b = 0; b <= i; ++b)          sum += av_sum[b * CCH + c];
        for (int b = 56 + i; b < Bsz; ++b)    sum += av_sum[b * CCH + c];
        AV[i * CCH + c] = sum * scale * (1.0f / 9.0f);
    } else {
        for (int b = i - WIN; b < i; ++b)     sum += av_sum[b * CCH + c];
        AV[i * CCH + c] = sum * scale * 0.125f;
    }
}

// ---------------------------------------------------------------------------
// One 16x16 f32 D-tile per wave via V_WMMA_F32_16X16X4_F32.
// A 16x4 layout: lanes 0-15 hold (K=0,K=1), lanes 16-31 hold (K=2,K=3).
// B 4x16 mirrors it (rows striped across lanes within a VGPR).
__device__ __forceinline__ v8f wmma_tile_f32(const float* __restrict__ A, int lda,
                                             const float* __restrict__ B, int ldb,
                                             int K, int lane) {
    v8f acc = {};
    const int r    = lane & 15;               // A row / B col within tile
    const int kofs = (lane >> 4) << 1;        // 0 or 2
    for (int k0 = 0; k0 < K; k0 += 4) {
        if (k0 + 4 < K)                       // global_prefetch_b8 next B rows
            __builtin_prefetch(&B[(size_t)(k0 + 4 + kofs) * ldb + r], 0, 1);
        v2f a, b;
        a.x = A[(size_t)r * lda + k0 + kofs];
        a.y = A[(size_t)r * lda + k0 + kofs + 1];
        b.x = B[(size_t)(k0 + kofs)     * ldb + r];
        b.y = B[(size_t)(k0 + kofs + 1) * ldb + r];
        acc = __builtin_amdgcn_wmma_f32_16x16x4_f32(
            /*neg_a=*/false, a, /*neg_b=*/false, b,
            /*c_mod=*/(short)0, acc, /*reuse_a=*/false, /*reuse_b=*/false);
    }
    return acc;
}

// layer 1: [64,256] @ [256,768] + bias, relu. Both paths. 384 waves.
__global__ void k_mlp1(const float* __restrict__ MX, const float* __restrict__ AV,
                       const float* __restrict__ W1max, const float* __restrict__ b1max,
                       const float* __restrict__ W1avg, const float* __restrict__ b1avg,
                       float* __restrict__ Hmax, float* __restrict__ Havg) {
    const int lane = threadIdx.x & 31;
    const int gw   = blockIdx.x * (blockDim.x >> 5) + (threadIdx.x >> 5);
    const int path = gw / 192;                // 4 Mtiles * 48 Ntiles per path
    const int t    = gw % 192;
    const int tm = t & 3, tn = t >> 2;

    const float* X  = path ? AV    : MX;
    const float* W  = path ? W1avg : W1max;
    const float* bb = path ? b1avg : b1max;
    float*       H  = path ? Havg  : Hmax;

    v8f acc = wmma_tile_f32(X + (size_t)tm * 16 * CCH, CCH,
                            W + (size_t)tn * 16, HID, CCH, lane);

    const int col   = tn * 16 + (lane & 15);
    const int rbase = tm * 16 + (lane >> 4) * 8;     // D: lanes>=16 hold M=8..15
    const float bias = bb[col];
    #pragma unroll
    for (int r = 0; r < 8; ++r) {
        float v = acc[r] + bias;
        H[(size_t)(rbase + r) * HID + col] = v > 0.0f ? v : 0.0f;
    }
}

// layer 2: [64,768] @ [768,256] + bias, relu. Both paths. 128 waves.
__global__ void k_mlp2(const float* __restrict__ Hmax, const float* __restrict__ Havg,
                       const float* __restrict__ W2max, const float* __restrict__ b2max,
                       const float* __restrict__ W2avg, const float* __restrict__ b2avg,
                       float* __restrict__ M2, float* __restrict__ A2) {
    const int lane = threadIdx.x & 31;
    const int gw   = blockIdx.x * (blockDim.x >> 5) + (threadIdx.x >> 5);
    const int path = gw >> 6;                 // 4 Mtiles * 16 Ntiles per path
    const int t    = gw & 63;
    const int tm = t & 3, tn = t >> 2;

    const float* X  = path ? Havg  : Hmax;
    const float* W  = path ? W2avg : W2max;
    const float* bb = path ? b2avg : b2max;
    float*       O  = path ? A2    : M2;

    v8f acc = wmma_tile_f32(X + (size_t)tm * 16 * HID, HID,
                            W + (size_t)tn * 16, CCH, HID, lane);

    const int col   = tn * 16 + (lane & 15);
    const int rbase = tm * 16 + (lane >> 4) * 8;
    const float bias = bb[col];
    #pragma unroll
    for (int r = 0; r < 8; ++r) {
        float v = acc[r] + bias;
        O[(size_t)(rbase + r) * CCH + col] = v > 0.0f ? v : 0.0f;
    }
}

// ---------------------------------------------------------------------------
// att[i,:] = softmax(sigmoid(M2[i,:] + A2[i,:]))   (block = one row)
__global__ void k_att(const float* __restrict__ M2, const float* __restrict__ A2,
                      float* __restrict__ att) {
    __shared__ float red[CCH];
    const int c = threadIdx.x;
    const int i = blockIdx.x;

    float z = M2[i * CCH + c] + A2[i * CCH + c];
    float s = 1.0f / (1.0f + expf(-z));

    red[c] = s;
    __syncthreads();
    for (int off = CCH / 2; off > 0; off >>= 1) {
        if (c < off) red[c] = fmaxf(red[c], red[c + off]);
        __syncthreads();
    }
    float mx = red[0];
    __syncthreads();

    float e = expf(s - mx);
    red[c] = e;
    __syncthreads();
    for (int off = CCH / 2; off > 0; off >>= 1) {
        if (c < off) red[c] += red[c + off];
        __syncthreads();
    }
    att[i * CCH + c] = e / red[0];
}

// final cyclic window mean over batch of att -> out [64,256]
__global__ void k_final(const float* __restrict__ att, float* __restrict__ out) {
    const int c = threadIdx.x;
    const int i = blockIdx.x;
    float sum = 0.0f;
    if (i < WIN) {
        for (int b = 0; b <= i; ++b)       sum += att[b * CCH + c];
        for (int b = 56 + i; b < Bsz; ++b) sum += att[b * CCH + c];
        out[i * CCH + c] = sum * (1.0f / 9.0f);
    } else {
        for (int b = i - WIN; b < i; ++b)  sum += att[b * CCH + c];
        out[i * CCH + c] = sum * 0.125f;
    }
}

// ---------------------------------------------------------------------------
extern "C" void kernel_launch(void* const* d_in, const int* in_sizes, int n_in,
                              void* d_out, int out_size, void* d_ws, size_t ws_size,
                              hipStream_t stream) {
    const float* x     = (const float*)d_in[0];
    const float* W1max = (const float*)d_in[1];
    const float* b1max = (const float*)d_in[2];
    const float* W2max = (const float*)d_in[3];
    const float* b2max = (const float*)d_in[4];
    const float* W1avg = (const float*)d_in[5];
    const float* b1avg = (const float*)d_in[6];
    const float* W2avg = (const float*)d_in[7];
    const float* b2avg = (const float*)d_in[8];
    // d_in[9] = windows (python int == 8, baked into WIN)

    float* ws = (float*)d_ws;                  // 163840 floats = 640 KB used
    unsigned* mx_enc = (unsigned*)ws;          // [64*256]
    float* av_sum = ws + 16384;                // [64*256]
    float* MX     = ws + 32768;                // [64*256]
    float* AV     = ws + 49152;                // [64*256]
    float* Hmax   = ws + 65536;                // [64*768]
    float* Havg   = ws + 114688;               // [64*768]
    float* M2     = ws + 0;                    // reuse mx_enc slot (done)
    float* A2     = ws + 16384;                // reuse av_sum slot (done)
    float* att    = ws + 32768;                // reuse MX slot (done)
    float* out    = (float*)d_out;

    k_init       <<<64,  256, 0, stream>>>(mx_enc, av_sum);
    k_window_pool<<<256, 256, Bsz * CCH * sizeof(float), stream>>>(x, mx_enc, av_sum);
    k_finalize   <<<64,  256, 0, stream>>>(mx_enc, av_sum, MX, AV);
    k_mlp1       <<<48,  256, 0, stream>>>(MX, AV, W1max, b1max, W1avg, b1avg, Hmax, Havg);
    k_mlp2       <<<16,  256, 0, stream>>>(Hmax, Havg, W2max, b2max, W2avg, b2avg, M2, A2);
    k_att        <<<64,  256, 0, stream>>>(M2, A2, att);
    k_final      <<<64,  256, 0, stream>>>(att, out);
}